// DynamicEmbedder_4D_offset_add_noise_78400333021726
// MI455X (gfx1250) — compile-verified
//
#include <hip/hip_runtime.h>
#include <hip/hip_bf16.h>

#define GRIDX 128
#define GRIDY 128
#define GRIDZ 8
#define NKEYS (GRIDX * GRIDY * GRIDZ)   // 131072
#define CAP   64
#define KSEL  16

typedef float v2f __attribute__((ext_vector_type(2)));
typedef float v8f __attribute__((ext_vector_type(8)));

__device__ __forceinline__ int voxkey(const int* __restrict__ v, int i) {
    // coors are (z, y, x): key = z*(Sx*Sy) + y*Sx + x
    return v[i * 3 + 0] * (GRIDX * GRIDY) + v[i * 3 + 1] * GRIDX + v[i * 3 + 2];
}

// ---------------- phase 1: zero histogram ----------------
__global__ void init_counts_kernel(int* __restrict__ counts) {
    int i = blockIdx.x * blockDim.x + threadIdx.x;
    if (i < NKEYS) counts[i] = 0;
}

// ---------------- phase 2: histogram dense keys ----------------
__global__ void hist_kernel(const int* __restrict__ dvox, int* __restrict__ counts, int M) {
    int j = blockIdx.x * blockDim.x + threadIdx.x;
    if (j >= M) return;
    atomicAdd(&counts[voxkey(dvox, j)], 1);
}

// ---------------- phase 3: exclusive scan (single block) ----------------
__global__ __launch_bounds__(1024) void scan_kernel(const int* __restrict__ counts,
                                                    int* __restrict__ offsets,
                                                    int* __restrict__ cursor) {
    __shared__ int lds[1024];
    __shared__ int s_base;
    const int tid = threadIdx.x;
    if (tid == 0) s_base = 0;
    __syncthreads();
    for (int chunk = 0; chunk < NKEYS / 1024; ++chunk) {
        int i = chunk * 1024 + tid;
        int v = counts[i];
        lds[tid] = v;
        __syncthreads();
        for (int off = 1; off < 1024; off <<= 1) {
            int t = (tid >= off) ? lds[tid - off] : 0;
            __syncthreads();
            lds[tid] += t;
            __syncthreads();
        }
        int excl = lds[tid] - v + s_base;
        offsets[i] = excl;
        cursor[i]  = excl;
        int tot = lds[1023];
        __syncthreads();
        if (tid == 0) s_base += tot;
        __syncthreads();
    }
}

// ---------------- phase 4: scatter dense indices ----------------
__global__ void scatter_kernel(const int* __restrict__ dvox, int* __restrict__ cursor,
                               int* __restrict__ sidx, int M) {
    int j = blockIdx.x * blockDim.x + threadIdx.x;
    if (j >= M) return;
    int pos = atomicAdd(&cursor[voxkey(dvox, j)], 1);
    sidx[pos] = j;
}

// ---------------- phase 5: per-bucket sort (restore stable order) ----------------
__global__ void bucket_sort_kernel(const int* __restrict__ offsets, const int* __restrict__ counts,
                                   int* __restrict__ sidx) {
    int b = blockIdx.x * blockDim.x + threadIdx.x;
    if (b >= NKEYS) return;
    int st = offsets[b], n = counts[b];
    for (int i = 1; i < n; ++i) {
        int v = sidx[st + i];
        int j = i - 1;
        while (j >= 0 && sidx[st + j] > v) { sidx[st + j + 1] = sidx[st + j]; --j; }
        sidx[st + j + 1] = v;
    }
}

// ---------------- phase 6: WMMA KNN, one wave per sparse point ----------------
__global__ __launch_bounds__(256) void knn_wmma_kernel(
    const float* __restrict__ sp, const int* __restrict__ svox,
    const int* __restrict__ offsets, const int* __restrict__ counts,
    const int* __restrict__ sidx, const float* __restrict__ dp,
    float* __restrict__ out, int N)
{
    __shared__ float s_cand[8][CAP * 3 + 4];  // candidate xyz per wave
    __shared__ float s_score[8][CAP];

    const int wave = threadIdx.x >> 5;
    const int lane = threadIdx.x & 31;
    int i = blockIdx.x * 8 + wave;
    const bool active = (i < N);
    if (!active) i = N - 1;  // clamp; redundant compute, stores masked later

    const float sx = sp[i * 3 + 0];
    const float sy = sp[i * 3 + 1];
    const float sz = sp[i * 3 + 2];
    const int key     = voxkey(svox, i);
    const int start   = offsets[key];
    const int cntTrue = counts[key];
    const int cnt     = cntTrue > CAP ? CAP : cntTrue;

    // Gather up to 64 candidates into LDS (stable, sorted bucket order).
    #pragma unroll
    for (int r = 0; r < 2; ++r) {
        int s0 = lane + r * 32;
        float cx = 0.f, cy = 0.f, cz = 0.f;
        if (s0 < cnt) {
            int di = sidx[start + s0];
            cx = dp[di * 3 + 0];
            cy = dp[di * 3 + 1];
            cz = dp[di * 3 + 2];
        }
        s_cand[wave][s0 * 3 + 0] = cx;
        s_cand[wave][s0 * 3 + 1] = cy;
        s_cand[wave][s0 * 3 + 2] = cz;
    }
    __syncthreads();

    // A (16x4 f32): every row = (sx, sy, sz, 1). Lanes<16 carry K0/K1, lanes>=16 K2/K3.
    const bool hi = (lane >= 16);
    v2f a;
    a.x = hi ? sz : sx;
    a.y = hi ? 1.0f : sy;

    // 4 WMMAs: score(col j) = -2*s.c_j + |c_j|^2  (|s|^2 is row-constant, irrelevant to ranking)
    #pragma unroll
    for (int g = 0; g < 4; ++g) {
        int j = g * 16 + (lane & 15);
        float cx = s_cand[wave][j * 3 + 0];
        float cy = s_cand[wave][j * 3 + 1];
        float cz = s_cand[wave][j * 3 + 2];
        v2f b;
        b.x = hi ? (-2.0f * cz) : (-2.0f * cx);
        b.y = hi ? (cx * cx + cy * cy + cz * cz) : (-2.0f * cy);
        v8f acc = {};
        acc = __builtin_amdgcn_wmma_f32_16x16x4_f32(false, a, false, b, (short)0, acc,
                                                    false, false);
        // D VGPR0: lane<16 -> row0 col (lane), lane>=16 -> row8 col (lane-16); rows identical.
        float sc = acc[0];
        s_score[wave][j] = (j < cnt) ? sc : __builtin_inff();  // both half-lanes write same value
    }
    __syncthreads();

    // 16 rounds of wave-wide argmin (tie -> lower slot == top_k tie-break).
    float sc0 = s_score[wave][lane];
    float sc1 = s_score[wave][lane + 32];
    int mySel = 0;
    for (int k = 0; k < KSEL; ++k) {
        float v; int s;
        if (sc1 < sc0) { v = sc1; s = lane + 32; } else { v = sc0; s = lane; }
        #pragma unroll
        for (int m = 16; m >= 1; m >>= 1) {
            float ov = __shfl_xor(v, m, 32);
            int   os = __shfl_xor(s, m, 32);
            if (ov < v || (ov == v && os < s)) { v = ov; s = os; }
        }
        if (s == lane)      sc0 = __builtin_inff();
        if (s == lane + 32) sc1 = __builtin_inff();
        if (lane == k) mySel = s;   // lane k owns selection k
    }

    // Cyclic repeat when bucket has < K points; NaN when empty.
    int src = lane;
    if (cntTrue > 0 && cntTrue < KSEL) src = (lane < cntTrue) ? lane : (lane % cntTrue);
    int slotFinal = __shfl(mySel, src, 32);

    if (active && lane < KSEL) {
        float ox, oy, oz;
        if (cntTrue == 0) {
            ox = oy = oz = __builtin_nanf("");
        } else {
            ox = s_cand[wave][slotFinal * 3 + 0];
            oy = s_cand[wave][slotFinal * 3 + 1];
            oz = s_cand[wave][slotFinal * 3 + 2];
        }
        size_t o = ((size_t)i * KSEL + lane) * 3;
        out[o + 0] = ox;
        out[o + 1] = oy;
        out[o + 2] = oz;
    }
}

extern "C" void kernel_launch(void* const* d_in, const int* in_sizes, int n_in,
                              void* d_out, int out_size, void* d_ws, size_t ws_size,
                              hipStream_t stream) {
    const float* sp   = (const float*)d_in[0];  // sparse_points (N,3)
    const float* dp   = (const float*)d_in[1];  // dense_points  (M,3)
    const int*   svox = (const int*)d_in[2];    // sparse_voxels (N,3) z,y,x
    const int*   dvox = (const int*)d_in[3];    // dense_voxels  (M,3) z,y,x
    const int N = in_sizes[0] / 3;
    const int M = in_sizes[1] / 3;

    int* counts  = (int*)d_ws;           // NKEYS
    int* offsets = counts + NKEYS;       // NKEYS
    int* cursor  = offsets + NKEYS;      // NKEYS
    int* sidx    = cursor + NKEYS;       // M
    float* out   = (float*)d_out;

    init_counts_kernel<<<NKEYS / 256, 256, 0, stream>>>(counts);
    hist_kernel<<<(M + 255) / 256, 256, 0, stream>>>(dvox, counts, M);
    scan_kernel<<<1, 1024, 0, stream>>>(counts, offsets, cursor);
    scatter_kernel<<<(M + 255) / 256, 256, 0, stream>>>(dvox, cursor, sidx, M);
    bucket_sort_kernel<<<NKEYS / 256, 256, 0, stream>>>(offsets, counts, sidx);
    knn_wmma_kernel<<<(N + 7) / 8, 256, 0, stream>>>(sp, svox, offsets, counts, sidx, dp,
                                                     out, N);
}